// GAT_30580167147784
// MI455X (gfx1250) — compile-verified
//
#include <hip/hip_runtime.h>
#include <hip/hip_bf16.h>

typedef float v2f __attribute__((ext_vector_type(2)));
typedef float v8f __attribute__((ext_vector_type(8)));

#define LN_EPS 1e-5f
#define NEG_SLOPE 0.2f

// ---------------------------------------------------------------------------
// WMMA f32 GEMM:  C[M x Ncols] = A[M x K] * B[Ncols x K]^T   (K = 128)
// One wave (32 threads) computes one 16x16 tile of C using V_WMMA_F32_16X16X4_F32.
// A-matrix lane layout (ISA 7.12.2, 32-bit 16x4): lane L<16 holds row M=L,
// K=k+0 (v0) and K=k+1 (v1); lanes 16..31 hold K=k+2 / k+3.  B symmetric.
// C/D layout: VGPR r -> row (r + 8*(lane>=16)), col = lane&15.
// ---------------------------------------------------------------------------
__global__ void gemm_wmma_nt(const float* __restrict__ A,
                             const float* __restrict__ B,
                             float* __restrict__ C,
                             int K, int Ncols) {
  const int tileM = blockIdx.x * 16;
  const int tileN = blockIdx.y * 16;
  const int lane  = threadIdx.x;          // 0..31
  const int mn    = lane & 15;
  const int khalf = (lane >> 4) * 2;      // 0 or 2

  const float* __restrict__ arow = A + (size_t)(tileM + mn) * K;
  const float* __restrict__ brow = B + (size_t)(tileN + mn) * K;

  v8f acc = {};
#pragma unroll 8
  for (int k = 0; k < K; k += 4) {
    __builtin_prefetch(arow + k + 32, 0, 0);   // global_prefetch_b8
    v2f a, b;
    a[0] = arow[k + khalf];
    a[1] = arow[k + khalf + 1];
    b[0] = brow[k + khalf];
    b[1] = brow[k + khalf + 1];
    acc = __builtin_amdgcn_wmma_f32_16x16x4_f32(
        /*neg_a=*/false, a, /*neg_b=*/false, b,
        /*c_mod=*/(short)0, acc, /*reuse_a=*/false, /*reuse_b=*/false);
  }

  const int rbase = tileM + ((lane >> 4) << 3);
#pragma unroll
  for (int r = 0; r < 8; ++r) {
    C[(size_t)(rbase + r) * Ncols + tileN + mn] = acc[r];
  }
}

// ---------------------------------------------------------------------------
// Per-node attention coefficients: as[n,h] = sum_c h[n,h,c]*a_src[h,c]
// ---------------------------------------------------------------------------
__global__ void node_alpha(const float* __restrict__ h,
                           const float* __restrict__ a_src,
                           const float* __restrict__ a_dst,
                           float* __restrict__ as_out,
                           float* __restrict__ ad_out,
                           int width, int cper) {
  const int n = blockIdx.x;
  const int c = threadIdx.x;               // width threads (128 or 256)
  __shared__ float ss[256];
  __shared__ float sd[256];
  float hv = h[(size_t)n * width + c];
  ss[c] = hv * a_src[c];
  sd[c] = hv * a_dst[c];
  __syncthreads();
  if ((c & (cper - 1)) == 0) {
    float s = 0.f, d = 0.f;
    for (int i = 0; i < cper; ++i) { s += ss[c + i]; d += sd[c + i]; }
    const int head = c / cper;
    as_out[(size_t)n * 4 + head] = s;
    ad_out[(size_t)n * 4 + head] = d;
  }
}

// Monotone float <-> uint order encoding (for atomic segment-max on f32)
__device__ __forceinline__ unsigned fenc(float f) {
  unsigned u = __float_as_uint(f);
  return (u & 0x80000000u) ? ~u : (u | 0x80000000u);
}
__device__ __forceinline__ float fdec(unsigned u) {
  unsigned b = (u & 0x80000000u) ? (u ^ 0x80000000u) : ~u;
  return __uint_as_float(b);
}

// e = leaky_relu(as[src] + ad[dst]); atomicMax per (dst, head)
__global__ void edge_logits(const int* __restrict__ src, const int* __restrict__ dst,
                            const float* __restrict__ as, const float* __restrict__ ad,
                            float* __restrict__ ebuf, unsigned* __restrict__ emax,
                            int total) {
  int idx = blockIdx.x * blockDim.x + threadIdx.x;
  if (idx >= total) return;
  const int e = idx >> 2, hh = idx & 3;
  const int s = src[e], d = dst[e];
  float v = as[(size_t)s * 4 + hh] + ad[(size_t)d * 4 + hh];
  v = (v > 0.f) ? v : NEG_SLOPE * v;
  ebuf[idx] = v;
  atomicMax(&emax[(size_t)d * 4 + hh], fenc(v));
}

// ex = exp(e - emax[dst]); denom[dst] += ex
__global__ void edge_exp(const int* __restrict__ dst,
                         float* __restrict__ ebuf,
                         const unsigned* __restrict__ emax,
                         float* __restrict__ denom, int total) {
  int idx = blockIdx.x * blockDim.x + threadIdx.x;
  if (idx >= total) return;
  const int e = idx >> 2, hh = idx & 3;
  const int d = dst[e];
  float m  = fdec(emax[(size_t)d * 4 + hh]);
  float ex = expf(ebuf[idx] - m);
  ebuf[idx] = ex;
  atomicAdd(&denom[(size_t)d * 4 + hh], ex);
}

// alpha = ex / (denom[dst] + 1e-16)   (in place; ebuf is the d_out alpha slot)
__global__ void edge_norm(const int* __restrict__ dst,
                          float* __restrict__ ebuf,
                          const float* __restrict__ denom, int total) {
  int idx = blockIdx.x * blockDim.x + threadIdx.x;
  if (idx >= total) return;
  const int e = idx >> 2, hh = idx & 3;
  ebuf[idx] = ebuf[idx] / (denom[(size_t)dst[e] * 4 + hh] + 1e-16f);
}

// agg[dst, c] += h[src, c] * alpha[e, head(c)]   (one block per edge)
__global__ void edge_scatter(const int* __restrict__ src, const int* __restrict__ dst,
                             const float* __restrict__ h,
                             const float* __restrict__ alpha,
                             float* __restrict__ agg,
                             int width, int cper) {
  const int e = blockIdx.x;
  const int c = threadIdx.x;               // width threads
  const int s = src[e], d = dst[e];
  const int head = c / cper;
  const float a = alpha[(size_t)e * 4 + head];
  atomicAdd(&agg[(size_t)d * width + c], h[(size_t)s * width + c] * a);
}

// out = LN(elu(agg + bias)) * lnw + lnb  [+ skip + bskip]   (width == 128)
__global__ void ep_ln(const float* __restrict__ agg, const float* __restrict__ bias,
                      const float* __restrict__ lnw, const float* __restrict__ lnb,
                      const float* __restrict__ skip, const float* __restrict__ bskip,
                      float* __restrict__ outp, int width) {
  const int n = blockIdx.x;
  const int c = threadIdx.x;
  __shared__ float sh[128];
  float v = agg[(size_t)n * width + c] + bias[c];
  v = (v > 0.f) ? v : expm1f(v);           // ELU
  sh[c] = v;
  __syncthreads();
  for (int off = width >> 1; off > 0; off >>= 1) {
    if (c < off) sh[c] += sh[c + off];
    __syncthreads();
  }
  const float mu = sh[0] / (float)width;
  __syncthreads();
  const float dv = v - mu;
  sh[c] = dv * dv;
  __syncthreads();
  for (int off = width >> 1; off > 0; off >>= 1) {
    if (c < off) sh[c] += sh[c + off];
    __syncthreads();
  }
  const float var = sh[0] / (float)width;
  float y = dv * rsqrtf(var + LN_EPS) * lnw[c] + lnb[c];
  if (skip) y += skip[(size_t)n * width + c] + bskip[c];
  outp[(size_t)n * width + c] = y;
}

// layer-3 head mean: out[n,c] = mean_h agg[n, h*64+c] + b3[c]
__global__ void ep_mean(const float* __restrict__ agg, const float* __restrict__ bias,
                        float* __restrict__ outp) {
  const int n = blockIdx.x;
  const int c = threadIdx.x;               // 64
  float s = 0.f;
#pragma unroll
  for (int hh = 0; hh < 4; ++hh) s += agg[(size_t)n * 256 + hh * 64 + c];
  outp[(size_t)n * 64 + c] = 0.25f * s + bias[c];
}

// ---------------------------------------------------------------------------
extern "C" void kernel_launch(void* const* d_in, const int* in_sizes, int n_in,
                              void* d_out, int out_size, void* d_ws, size_t ws_size,
                              hipStream_t stream) {
  const int IN = 128, D1 = 128;
  const int N = in_sizes[0] / IN;          // 50000
  const int E = in_sizes[1] / 2;           // 800000

  const float* x     = (const float*)d_in[0];
  const int*   ei    = (const int*)d_in[1];
  const int*   src   = ei;
  const int*   dst   = ei + E;
  const float* W1    = (const float*)d_in[2];
  const float* as1   = (const float*)d_in[3];
  const float* ad1   = (const float*)d_in[4];
  const float* b1    = (const float*)d_in[5];
  const float* Wsk   = (const float*)d_in[6];
  const float* bsk   = (const float*)d_in[7];
  const float* ln1w  = (const float*)d_in[8];
  const float* ln1b  = (const float*)d_in[9];
  const float* W2    = (const float*)d_in[10];
  const float* as2   = (const float*)d_in[11];
  const float* ad2   = (const float*)d_in[12];
  const float* b2    = (const float*)d_in[13];
  const float* ln2w  = (const float*)d_in[14];
  const float* ln2b  = (const float*)d_in[15];
  const float* W3    = (const float*)d_in[16];
  const float* as3   = (const float*)d_in[17];
  const float* ad3   = (const float*)d_in[18];
  const float* b3    = (const float*)d_in[19];

  float* out_h3 = (float*)d_out;
  float* alpha1 = out_h3 + (size_t)N * 64;
  float* alpha2 = alpha1 + (size_t)E * 4;
  float* alpha3 = alpha2 + (size_t)E * 4;

  // workspace carve-out
  float* p       = (float*)d_ws;
  float* hbuf    = p; p += (size_t)N * 256;   // projected features (max width 256)
  float* aggbuf  = p; p += (size_t)N * 256;   // scatter accumulator
  float* featbuf = p; p += (size_t)N * 128;   // layer inputs h1 / h2
  float* skipbuf = p; p += (size_t)N * 128;   // x @ Wskip^T
  float* asb     = p; p += (size_t)N * 4;
  float* adb     = p; p += (size_t)N * 4;
  float* denom   = p; p += (size_t)N * 4;
  unsigned* emax = (unsigned*)p;              // N*4

  const int totEH   = E * 4;
  const int ethread = 256;
  const int eblocks = (totEH + ethread - 1) / ethread;
  const dim3 w32(32, 1, 1);

  // ---------------- Layer 1 ----------------
  gemm_wmma_nt<<<dim3(N / 16, D1 / 16), w32, 0, stream>>>(x, W1, hbuf, IN, D1);
  gemm_wmma_nt<<<dim3(N / 16, D1 / 16), w32, 0, stream>>>(x, Wsk, skipbuf, IN, D1);
  node_alpha<<<N, 128, 0, stream>>>(hbuf, as1, ad1, asb, adb, 128, 32);
  hipMemsetAsync(emax,   0, (size_t)N * 4 * sizeof(unsigned), stream);
  hipMemsetAsync(denom,  0, (size_t)N * 4 * sizeof(float), stream);
  hipMemsetAsync(aggbuf, 0, (size_t)N * 128 * sizeof(float), stream);
  edge_logits<<<eblocks, ethread, 0, stream>>>(src, dst, asb, adb, alpha1, emax, totEH);
  edge_exp   <<<eblocks, ethread, 0, stream>>>(dst, alpha1, emax, denom, totEH);
  edge_norm  <<<eblocks, ethread, 0, stream>>>(dst, alpha1, denom, totEH);
  edge_scatter<<<E, 128, 0, stream>>>(src, dst, hbuf, alpha1, aggbuf, 128, 32);
  ep_ln<<<N, 128, 0, stream>>>(aggbuf, b1, ln1w, ln1b, skipbuf, bsk, featbuf, 128);

  // ---------------- Layer 2 ----------------
  gemm_wmma_nt<<<dim3(N / 16, D1 / 16), w32, 0, stream>>>(featbuf, W2, hbuf, D1, D1);
  node_alpha<<<N, 128, 0, stream>>>(hbuf, as2, ad2, asb, adb, 128, 32);
  hipMemsetAsync(emax,   0, (size_t)N * 4 * sizeof(unsigned), stream);
  hipMemsetAsync(denom,  0, (size_t)N * 4 * sizeof(float), stream);
  hipMemsetAsync(aggbuf, 0, (size_t)N * 128 * sizeof(float), stream);
  edge_logits<<<eblocks, ethread, 0, stream>>>(src, dst, asb, adb, alpha2, emax, totEH);
  edge_exp   <<<eblocks, ethread, 0, stream>>>(dst, alpha2, emax, denom, totEH);
  edge_norm  <<<eblocks, ethread, 0, stream>>>(dst, alpha2, denom, totEH);
  edge_scatter<<<E, 128, 0, stream>>>(src, dst, hbuf, alpha2, aggbuf, 128, 32);
  ep_ln<<<N, 128, 0, stream>>>(aggbuf, b2, ln2w, ln2b, nullptr, nullptr, featbuf, 128);

  // ---------------- Layer 3 (H=4, C=64, concat=False) ----------------
  gemm_wmma_nt<<<dim3(N / 16, 256 / 16), w32, 0, stream>>>(featbuf, W3, hbuf, D1, 256);
  node_alpha<<<N, 256, 0, stream>>>(hbuf, as3, ad3, asb, adb, 256, 64);
  hipMemsetAsync(emax,   0, (size_t)N * 4 * sizeof(unsigned), stream);
  hipMemsetAsync(denom,  0, (size_t)N * 4 * sizeof(float), stream);
  hipMemsetAsync(aggbuf, 0, (size_t)N * 256 * sizeof(float), stream);
  edge_logits<<<eblocks, ethread, 0, stream>>>(src, dst, asb, adb, alpha3, emax, totEH);
  edge_exp   <<<eblocks, ethread, 0, stream>>>(dst, alpha3, emax, denom, totEH);
  edge_norm  <<<eblocks, ethread, 0, stream>>>(dst, alpha3, denom, totEH);
  edge_scatter<<<E, 256, 0, stream>>>(src, dst, hbuf, alpha3, aggbuf, 256, 64);
  ep_mean<<<N, 64, 0, stream>>>(aggbuf, b3, out_h3);
}